// DechunkingLayer_32839319945812
// MI455X (gfx1250) — compile-verified
//
#include <hip/hip_runtime.h>

typedef float v4f   __attribute__((ext_vector_type(4)));
typedef int   i32x4 __attribute__((vector_size(16)));   // matches async builtin param type

#define T_     4096
#define LZ_    1024
#define D_     1024
#define TCHUNK 4          // timesteps per block
#define NROWS  (TCHUNK + 1)

#if defined(__HIP_DEVICE_COMPILE__) && __has_builtin(__builtin_amdgcn_global_load_async_to_lds_b128)
#  define USE_ASYNC 1
#else
#  define USE_ASYNC 0
#endif

__device__ __forceinline__ void wait_async_zero() {
#if defined(__HIP_DEVICE_COMPILE__)
#  if __has_builtin(__builtin_amdgcn_s_wait_asynccnt)
  __builtin_amdgcn_s_wait_asynccnt(0);
#  else
  asm volatile("s_wait_asynccnt 0x0" ::: "memory");
#  endif
  asm volatile("" ::: "memory");
#endif
}

// ---------------------------------------------------------------------
// Main kernel: gather + lerp, 4 timesteps per block.
// Block (batch, t0..t0+3) stages the 5 needed z-rows (idx[t0-1..t0+3])
// into LDS via CDNA5 async global->LDS B128 loads (ASYNCcnt), then each
// thread lerps one float4 per timestep and writes with non-temporal
// 128-bit stores (keeps the 32MB z array resident in the 192MB L2 while
// streaming the 128MB output).
// ---------------------------------------------------------------------
__global__ __launch_bounds__(256) void dechunk_lerp_kernel(const float* __restrict__ z,
                                                           const float* __restrict__ p,
                                                           const int* __restrict__ idx,
                                                           float* __restrict__ out) {
  const int blk   = blockIdx.x;
  const int batch = blk >> 10;                 // / (T_/TCHUNK) == /1024
  const int t0    = (blk & 1023) * TCHUNK;
  const int tid   = threadIdx.x;
  const size_t bofs = (size_t)batch * T_;

  // Block-uniform gather indices for rows t0-1 .. t0+3 (t0-1 aliased to t0 at batch start)
  int rix[NROWS];
  rix[0] = (t0 > 0) ? idx[bofs + t0 - 1] : idx[bofs + t0];
#pragma unroll
  for (int k = 0; k < TCHUNK; ++k) rix[k + 1] = idx[bofs + t0 + k];

#if USE_ASYNC
  __shared__ float sm[NROWS * D_];             // 5 rows x 4KB = 20KB
#pragma unroll
  for (int j = 0; j < NROWS; ++j) {
    const float* rp = z + ((size_t)batch * LZ_ + (size_t)rix[j]) * D_ + 4 * tid;
    __builtin_amdgcn_global_load_async_to_lds_b128(
        (__attribute__((address_space(1))) i32x4*)rp,
        (__attribute__((address_space(3))) i32x4*)(sm + j * D_ + 4 * tid), 0, 0);
  }
  wait_async_zero();
#endif

  const float coef0 = (t0 > 0) ? p[bofs + t0] : 1.0f;
#pragma unroll
  for (int k = 0; k < TCHUNK; ++k) {
    const float coef = (k == 0) ? coef0 : p[bofs + t0 + k];
#if USE_ASYNC
    v4f a = *(const v4f*)(sm + (k + 1) * D_ + 4 * tid);   // row at t
    v4f c = *(const v4f*)(sm + k * D_ + 4 * tid);         // row at t-1
#else
    v4f a = *(const v4f*)(z + ((size_t)batch * LZ_ + (size_t)rix[k + 1]) * D_ + 4 * tid);
    v4f c = *(const v4f*)(z + ((size_t)batch * LZ_ + (size_t)rix[k]) * D_ + 4 * tid);
#endif
    v4f r = coef * a + (1.0f - coef) * c;
    __builtin_nontemporal_store(r, (v4f*)(out + (bofs + t0 + k) * D_ + 4 * tid));
  }
}

// ---------------------------------------------------------------------
// Prefix-sum kernel: per-batch exclusive cumsum of gate b, clipped to
// [0, LZ-1]. One block per batch, 1024 threads, 4 elements per thread.
// ---------------------------------------------------------------------
__global__ __launch_bounds__(1024) void dechunk_scan_kernel(const int* __restrict__ bmat,
                                                            int* __restrict__ idx) {
  __shared__ int s[1024];
  const int batch = blockIdx.x;
  const int j = threadIdx.x;
  const int* bb = bmat + (size_t)batch * T_ + 4 * j;
  const int v0 = bb[0], v1 = bb[1], v2 = bb[2], v3 = bb[3];
  const int tot = v0 + v1 + v2 + v3;
  s[j] = tot;
  __syncthreads();
  // Hillis-Steele inclusive scan of per-thread totals
  for (int off = 1; off < 1024; off <<= 1) {
    int t = (j >= off) ? s[j - off] : 0;
    __syncthreads();
    s[j] += t;
    __syncthreads();
  }
  int run = s[j] - tot;  // exclusive prefix before this thread's 4 elements
  int* ii = idx + (size_t)batch * T_ + 4 * j;
  int c;
  c = run; c = c < 0 ? 0 : c; c = c > (LZ_ - 1) ? (LZ_ - 1) : c; ii[0] = c; run += v0;
  c = run; c = c < 0 ? 0 : c; c = c > (LZ_ - 1) ? (LZ_ - 1) : c; ii[1] = c; run += v1;
  c = run; c = c < 0 ? 0 : c; c = c > (LZ_ - 1) ? (LZ_ - 1) : c; ii[2] = c; run += v2;
  c = run; c = c < 0 ? 0 : c; c = c > (LZ_ - 1) ? (LZ_ - 1) : c; ii[3] = c;
}

// ---------------------------------------------------------------------
extern "C" void kernel_launch(void* const* d_in, const int* in_sizes, int n_in,
                              void* d_out, int out_size, void* d_ws, size_t ws_size,
                              hipStream_t stream) {
  const float* z = (const float*)d_in[0];
  const float* p = (const float*)d_in[1];
  const int*   b = (const int*)d_in[2];
  // d_in[3] = original_len (scalar), unused: shapes are fixed by the inputs.

  const int B = in_sizes[1] / T_;       // p is (B, T)
  int* idx = (int*)d_ws;                // B*T*4 = 128KB scratch

  dechunk_scan_kernel<<<dim3(B), dim3(1024), 0, stream>>>(b, idx);
  dechunk_lerp_kernel<<<dim3(B * (T_ / TCHUNK)), dim3(256), 0, stream>>>(z, p, idx,
                                                                         (float*)d_out);
}